// BasicRecursiveNN_63857573757109
// MI455X (gfx1250) — compile-verified
//
#include <hip/hip_runtime.h>
#include <hip/hip_bf16.h>

typedef float v2f __attribute__((ext_vector_type(2)));
typedef float v8f __attribute__((ext_vector_type(8)));

#define DD 512            // hidden size
#define KK 1024           // 2*DD (concat of two children)
#define XS_STRIDE 516     // 512 + 4 pad: bank-conflict-free, 16B-aligned rows
#define MAXLVL 48

// ---------------------------------------------------------------------------
// Kernel 1: leaf embedding gather  states[i, :] = word_emb[tokens[i], :]
// ---------------------------------------------------------------------------
__global__ void leaf_embed_kernel(const int* __restrict__ tokens,
                                  const float* __restrict__ word_emb,
                                  float* __restrict__ states, int L) {
    int idx = blockIdx.x * blockDim.x + threadIdx.x;   // one float4 each
    int total = L * (DD / 4);
    if (idx >= total) return;
    int leaf = idx / (DD / 4);
    int c4   = idx % (DD / 4);
    const float4* src = (const float4*)(word_emb + (size_t)tokens[leaf] * DD);
    float4*       dst = (float4*)(states + (size_t)leaf * DD);
    dst[c4] = src[c4];
}

// ---------------------------------------------------------------------------
// Kernel 2: level scheduling. Single block. lev[k] = 1 + max(lev children).
// Buckets internal-node creation indices into per-level compacted lists.
// ---------------------------------------------------------------------------
__global__ void schedule_kernel(const int* __restrict__ left,
                                const int* __restrict__ right, int L,
                                int* __restrict__ lev, int* __restrict__ order,
                                int* __restrict__ levelStart,
                                int* __restrict__ cursor) {
    int nInt = L - 1;
    int tid = threadIdx.x;
    for (int k = tid; k < nInt; k += blockDim.x) lev[k] = -1;
    for (int d = tid; d <= MAXLVL; d += blockDim.x) { levelStart[d] = 0; cursor[d] = 0; }
    __syncthreads();

    // Fixed-point sweeps; depth of balanced tree over 8192 leaves is 13 << 48.
    for (int sweep = 0; sweep < MAXLVL; ++sweep) {
        for (int k = tid; k < nInt; k += blockDim.x) {
            if (lev[k] < 0) {
                int l = left[k], r = right[k];
                int ll = (l < L) ? 0 : lev[l - L];
                int rr = (r < L) ? 0 : lev[r - L];
                if (ll >= 0 && rr >= 0) lev[k] = (ll > rr ? ll : rr) + 1;
            }
        }
        __syncthreads();
    }
    // Histogram levels.
    for (int k = tid; k < nInt; k += blockDim.x) atomicAdd(&cursor[lev[k]], 1);
    __syncthreads();
    if (tid == 0) {
        int acc = 0;
        for (int d = 0; d < MAXLVL; ++d) { levelStart[d] = acc; acc += cursor[d]; }
        levelStart[MAXLVL] = acc;
    }
    __syncthreads();
    for (int d = tid; d < MAXLVL; d += blockDim.x) cursor[d] = levelStart[d];
    __syncthreads();
    // Scatter node ids grouped by level (intra-level order irrelevant).
    for (int k = tid; k < nInt; k += blockDim.x) {
        int pos = atomicAdd(&cursor[lev[k]], 1);
        order[pos] = k;
    }
}

// ---------------------------------------------------------------------------
// Kernel 3: one tree level as a batched GEMM + tanh.
//   For each node tile (16 nodes) each wave computes a 16x16 feature tile of
//   H = tanh([Xl Xr] * W^T + b) with V_WMMA_F32_16X16X4_F32.
// grid = (64, 4), block = 256 (8 waves). wave w handles feature tile
// f = blockIdx.y*8 + w  (32 tiles of 16 features = 512).
// ---------------------------------------------------------------------------
__global__ void __launch_bounds__(256)
tree_level_kernel(float* __restrict__ states,
                  const float* __restrict__ fc_w,   // (512, 1024) row-major
                  const float* __restrict__ fc_b,   // (512,)
                  const int* __restrict__ order,
                  const int* __restrict__ left,
                  const int* __restrict__ right,
                  const int* __restrict__ levelStart,
                  int L, int level) {
    __shared__ float xs[16][XS_STRIDE];   // one child-half of X for 16 nodes
    __shared__ int   nodeIdxS[16];

    int start = levelStart[level];
    int count = levelStart[level + 1] - start;
    if (count <= 0) return;
    int nTiles = (count + 15) >> 4;

    int lane = threadIdx.x & 31;
    int wave = threadIdx.x >> 5;
    int f    = blockIdx.y * 8 + wave;     // 0..31
    int n0   = f * 16;
    int lanelo = lane & 15;               // A row / B+C column within tile
    int khalf  = (lane >> 4) << 1;        // 0 for lanes 0-15, 2 for lanes 16-31
    const float* wrow = fc_w + (size_t)(n0 + lanelo) * KK;  // W^T column

    for (int tile = blockIdx.x; tile < nTiles; tile += gridDim.x) {
        int base = start + (tile << 4);
        int rows = count - (tile << 4);
        if (rows > 16) rows = 16;

        if (threadIdx.x < 16)
            nodeIdxS[threadIdx.x] = (threadIdx.x < rows) ? order[base + threadIdx.x] : -1;
        __syncthreads();

        v8f c = {0.f, 0.f, 0.f, 0.f, 0.f, 0.f, 0.f, 0.f};

        // Two halves of K: h=0 -> left-child states (k 0..511),
        //                  h=1 -> right-child states (k 512..1023).
        for (int h = 0; h < 2; ++h) {
            // Stage 16 rows x 512 floats = 16 x 128 float4 (256 threads, 8 ea).
            for (int t = threadIdx.x; t < 16 * (DD / 4); t += blockDim.x) {
                int row = t >> 7;                 // /128
                int c4  = t & 127;
                float4 v = make_float4(0.f, 0.f, 0.f, 0.f);
                int nk = nodeIdxS[row];
                if (nk >= 0) {
                    int child = h ? right[nk] : left[nk];
                    v = ((const float4*)(states + (size_t)child * DD))[c4];
                }
                *(float4*)&xs[row][c4 * 4] = v;
            }
            __syncthreads();

            const float* wh = wrow + h * DD;
            #pragma unroll 8
            for (int k0 = 0; k0 < DD; k0 += 4) {
                // A fragment (16x4 f32): lane<16 -> K=k0,k0+1 ; lane>=16 -> K=k0+2,k0+3
                v2f a = *(const v2f*)&xs[lanelo][k0 + khalf];
                // B fragment (4x16 f32) from W^T, same K split, N = n0+lanelo.
                v2f b = *(const v2f*)(wh + k0 + khalf);
                c = __builtin_amdgcn_wmma_f32_16x16x4_f32(
                        /*neg_a=*/false, a, /*neg_b=*/false, b,
                        /*c_mod=*/(short)0, c,
                        /*reuse_a=*/false, /*reuse_b=*/false);
            }
            __syncthreads();   // before restaging xs
        }

        // Epilogue: C layout -> lane holds (M = r + 8*(lane>=16), N = lanelo).
        float bias = fc_b[n0 + lanelo];
        int mbase = (lane >> 4) << 3;
        #pragma unroll
        for (int r = 0; r < 8; ++r) {
            int m = mbase + r;
            int nk = nodeIdxS[m];
            if (m < rows && nk >= 0) {
                float hval = tanhf(c[r] + bias);
                states[(size_t)(L + nk) * DD + n0 + lanelo] = hval;
            }
        }
        __syncthreads();   // before nodeIdxS is overwritten next tile
    }
}

// ---------------------------------------------------------------------------
// Kernel 4: out[j,:] = states[postorder[j],:], plus final_state = root row.
// ---------------------------------------------------------------------------
__global__ void gather_kernel(const float* __restrict__ states,
                              const int* __restrict__ postorder,
                              float* __restrict__ out, int N) {
    long idx = (long)blockIdx.x * blockDim.x + threadIdx.x;  // one float4
    long total = (long)(N + 1) * (DD / 4);
    if (idx >= total) return;
    int row = (int)(idx / (DD / 4));
    int c4  = (int)(idx % (DD / 4));
    int src = (row < N) ? postorder[row] : postorder[N - 1];
    ((float4*)out)[(size_t)row * (DD / 4) + c4] =
        ((const float4*)(states + (size_t)src * DD))[c4];
}

// ---------------------------------------------------------------------------
extern "C" void kernel_launch(void* const* d_in, const int* in_sizes, int n_in,
                              void* d_out, int out_size, void* d_ws, size_t ws_size,
                              hipStream_t stream) {
    const int*   tokens    = (const int*)d_in[0];
    const int*   left_idx  = (const int*)d_in[1];
    const int*   right_idx = (const int*)d_in[2];
    const int*   postorder = (const int*)d_in[3];
    const float* word_emb  = (const float*)d_in[4];
    const float* fc_w      = (const float*)d_in[5];
    const float* fc_b      = (const float*)d_in[6];
    float* out = (float*)d_out;

    int L = in_sizes[0];
    int N = 2 * L - 1;

    // Workspace layout: states (N x 512 f32), then scheduler scratch.
    float* states = (float*)d_ws;
    size_t statesBytes = (size_t)N * DD * sizeof(float);
    char* p = (char*)d_ws + ((statesBytes + 255) & ~(size_t)255);
    int* lev        = (int*)p;  p += sizeof(int) * (size_t)L;
    int* order      = (int*)p;  p += sizeof(int) * (size_t)L;
    int* levelStart = (int*)p;  p += sizeof(int) * (MAXLVL + 1);
    int* cursor     = (int*)p;  p += sizeof(int) * (MAXLVL + 1);
    (void)ws_size; (void)n_in; (void)out_size;

    // 1) leaves
    {
        int total = L * (DD / 4);
        leaf_embed_kernel<<<(total + 255) / 256, 256, 0, stream>>>(
            tokens, word_emb, states, L);
    }
    // 2) schedule levels
    schedule_kernel<<<1, 256, 0, stream>>>(left_idx, right_idx, L,
                                           lev, order, levelStart, cursor);
    // 3) level-wise batched GEMM + tanh. Depth of balanced tree = ceil(log2 L).
    int depth = 0;
    while ((1 << depth) < L) ++depth;
    for (int d = 1; d <= depth; ++d) {
        tree_level_kernel<<<dim3(64, 4), 256, 0, stream>>>(
            states, fc_w, fc_b, order, left_idx, right_idx, levelStart, L, d);
    }
    // 4) postorder gather + final_state
    {
        long total = (long)(N + 1) * (DD / 4);
        gather_kernel<<<(int)((total + 255) / 256), 256, 0, stream>>>(
            states, postorder, out, N);
    }
}